// VariableLengthRecurrent_5738076308256
// MI455X (gfx1250) — compile-verified
//
#include <hip/hip_runtime.h>
#include <math.h>

typedef __attribute__((ext_vector_type(2))) float v2f;
typedef __attribute__((ext_vector_type(8))) float v8f;

#define HIDDEN 2048
#define INSZ   2048
#define BATCH  128
#define TSTEPS 128
#define KC     64

// ---- prefix sums of batch_size_per_step (device side; host can't read bsps) ----
__global__ void prefix_kernel(const int* __restrict__ bsps, int* __restrict__ prefix) {
    if (threadIdx.x == 0) {
        int c = 0;
        for (int t = 0; t < TSTEPS; ++t) { prefix[t] = c; c += bsps[t]; }
        prefix[TSTEPS] = c;
    }
}

__global__ void copy_kernel(const float* __restrict__ src, float* __restrict__ dst, int n) {
    int i = blockIdx.x * blockDim.x + threadIdx.x;
    if (i < n) dst[i] = src[i];
}

// One RNN timestep: hnext = tanh(x_t @ W_ih^T + hprev @ W_hh^T + b_ih + b_hh) for
// active rows (< bsps[t]); frozen rows are carried over. Active rows also write the
// packed output at row prefix[t]+b.
// Block = 128 threads = 4 waves; block tile = 16 rows x 128 cols.
// Each wave owns a 16x32 tile = two 16x16 accumulators sharing every A fragment.
__global__ __launch_bounds__(128) void rnn_step(
    const float* __restrict__ x,     const float* __restrict__ W_ih,
    const float* __restrict__ W_hh,  const float* __restrict__ b_ih,
    const float* __restrict__ b_hh,  const int*   __restrict__ bsps,
    const int*   __restrict__ prefix,const float* __restrict__ hprev,
    float* __restrict__ hnext,       float* __restrict__ outp, int t)
{
    __shared__ float As[16][KC + 1];        // +1 pad: column reads hit distinct banks
    __shared__ float Bs[4][2][KC][17];      // per wave: two 16-col K-major sub-tiles

    const int bt   = bsps[t];
    const int off  = prefix[t];
    const int base = blockIdx.y * 16;       // row-tile origin in the batch
    const int tid  = threadIdx.x;
    const int wave = tid >> 5;              // wave32
    const int lane = tid & 31;
    const int colbase = blockIdx.x * 128 + wave * 32;

    if (base >= bt) {
        // Entire tile frozen this step: carry hidden state forward for our columns.
        for (int i = tid; i < 16 * 128; i += 128) {
            int r = i >> 7, c = i & 127;
            size_t idx = (size_t)(base + r) * HIDDEN + blockIdx.x * 128 + c;
            hnext[idx] = hprev[idx];
        }
        return;
    }

    v8f acc0 = {};
    v8f acc1 = {};
    const int m     = lane & 15;            // row (A) / col (B,D) within 16
    const int ko    = (lane >> 4) << 1;     // K sub-offset: 0 (lanes 0-15), 2 (16-31)
    const int sub   = lane >> 4;            // which 16-col sub-tile this lane stages
    const int jj    = lane & 15;            // staged column within sub-tile

    #pragma unroll 1
    for (int phase = 0; phase < 2; ++phase) {
        const float* W = phase ? W_hh : W_ih;
        #pragma unroll 1
        for (int k0 = 0; k0 < 2048; k0 += KC) {
            __syncthreads();
            // Stage A tile (16 x KC) block-wide, float4 per thread x2.
            for (int i = tid; i < 16 * (KC / 4); i += 128) {
                int r = i >> 4, kq = (i & 15) * 4;
                float4 v;
                if (phase == 0) {
                    int row = base + r;
                    v = (row < bt)
                        ? *(const float4*)(x + (size_t)(off + row) * INSZ + k0 + kq)
                        : make_float4(0.f, 0.f, 0.f, 0.f);
                } else {
                    v = *(const float4*)(hprev + (size_t)(base + r) * HIDDEN + k0 + kq);
                }
                As[r][kq + 0] = v.x; As[r][kq + 1] = v.y;
                As[r][kq + 2] = v.z; As[r][kq + 3] = v.w;
            }
            // Stage this wave's two B sub-tiles: B[k][j] = W[col][k], one column/lane,
            // 16 x float4 contiguous along k (full cachelines per lane, L2-resident W).
            {
                const float* brow =
                    W + (size_t)(colbase + sub * 16 + jj) * 2048 + k0;
                #pragma unroll
                for (int i4 = 0; i4 < KC; i4 += 4) {
                    float4 v = *(const float4*)(brow + i4);
                    Bs[wave][sub][i4 + 0][jj] = v.x;
                    Bs[wave][sub][i4 + 1][jj] = v.y;
                    Bs[wave][sub][i4 + 2][jj] = v.z;
                    Bs[wave][sub][i4 + 3][jj] = v.w;
                }
            }
            __syncthreads();
            // 16 K-steps; each A fragment feeds two independent WMMAs.
            #pragma unroll
            for (int kk = 0; kk < KC; kk += 4) {
                v2f a, b0, b1;
                a.x  = As[m][kk + ko];
                a.y  = As[m][kk + ko + 1];
                b0.x = Bs[wave][0][kk + ko][m];
                b0.y = Bs[wave][0][kk + ko + 1][m];
                b1.x = Bs[wave][1][kk + ko][m];
                b1.y = Bs[wave][1][kk + ko + 1][m];
                acc0 = __builtin_amdgcn_wmma_f32_16x16x4_f32(
                        false, a, false, b0, (short)0, acc0, false, false);
                acc1 = __builtin_amdgcn_wmma_f32_16x16x4_f32(
                        false, a, false, b1, (short)0, acc1, false, false);
            }
        }
    }

    // Epilogue: bias + tanh, masked write of h and packed output; frozen rows carry.
    #pragma unroll
    for (int v = 0; v < 8; ++v) {
        int row  = base + v + ((lane >> 4) << 3);  // D layout: VGPR v -> row v / v+8
        int col0 = colbase + m;
        int col1 = colbase + 16 + m;
        float hn0 = tanhf(acc0[v] + b_ih[col0] + b_hh[col0]);
        float hn1 = tanhf(acc1[v] + b_ih[col1] + b_hh[col1]);
        size_t hi0 = (size_t)row * HIDDEN + col0;
        size_t hi1 = (size_t)row * HIDDEN + col1;
        if (row < bt) {
            hnext[hi0] = hn0;
            hnext[hi1] = hn1;
            outp[(size_t)(off + row) * HIDDEN + col0] = hn0;
            outp[(size_t)(off + row) * HIDDEN + col1] = hn1;
        } else {
            hnext[hi0] = hprev[hi0];
            hnext[hi1] = hprev[hi1];
        }
    }
}

extern "C" void kernel_launch(void* const* d_in, const int* in_sizes, int n_in,
                              void* d_out, int out_size, void* d_ws, size_t ws_size,
                              hipStream_t stream) {
    const float* x      = (const float*)d_in[0];
    const float* hidden = (const float*)d_in[1];
    const float* W_ih   = (const float*)d_in[2];
    const float* W_hh   = (const float*)d_in[3];
    const float* b_ih   = (const float*)d_in[4];
    const float* b_hh   = (const float*)d_in[5];
    const int*   bsps   = (const int*)d_in[6];

    float* h_final    = (float*)d_out;                          // [B, HIDDEN]
    float* out_packed = (float*)d_out + (size_t)BATCH * HIDDEN; // [total, HIDDEN]

    // Workspace: prefix sums + double-buffered hidden state (2 MB + 4 KB).
    int*   prefix = (int*)d_ws;
    float* hbuf0  = (float*)((char*)d_ws + 4096);
    float* hbuf1  = hbuf0 + (size_t)BATCH * HIDDEN;

    prefix_kernel<<<1, 32, 0, stream>>>(bsps, prefix);

    const int n = BATCH * HIDDEN;
    copy_kernel<<<(n + 255) / 256, 256, 0, stream>>>(hidden, hbuf0, n);

    dim3 grid(HIDDEN / 128, BATCH / 16);   // 16 col-groups x 8 row-tiles
    for (int t = 0; t < TSTEPS; ++t) {
        const float* hp = (t & 1) ? hbuf1 : hbuf0;
        float*       hn = (t & 1) ? hbuf0 : hbuf1;
        rnn_step<<<grid, 128, 0, stream>>>(x, W_ih, W_hh, b_ih, b_hh,
                                           bsps, prefix, hp, hn, out_packed, t);
    }
    // TSTEPS = 128 (even): final hidden state lands back in hbuf0.
    copy_kernel<<<(n + 255) / 256, 256, 0, stream>>>(hbuf0, h_final, n);
}